// RGCN_LinkPredictor_31396210933901
// MI455X (gfx1250) — compile-verified
//
#include <hip/hip_runtime.h>

typedef float v2f __attribute__((ext_vector_type(2)));
typedef float v4f __attribute__((ext_vector_type(4)));
typedef float v8f __attribute__((ext_vector_type(8)));

#define NNODES 100000
#define NEDGES 1600000
#define NPRED  200000
#define INC 128
#define HIDC 64
#define OUTC 32
#define NREL 3

// ---------------------------------------------------------------------------
// Dense GEMM via V_WMMA_F32_16X16X4_F32.
// grid.y = 0..2 -> Y[r] = X @ Wrel[r]         (no bias)
// grid.y = 3    -> H    = X @ Wroot + bias
// Each block: 8 waves; each wave owns one 16-row tile and NC/16 accum tiles.
// W is staged in LDS pair-interleaved: (k,col),(k+1,col) adjacent, so each
// B fragment is a single aligned ds_load_b64 into a contiguous VGPR pair.
// ---------------------------------------------------------------------------
template <int K, int NC>
__global__ __launch_bounds__(256) void gemm_wmma_kernel(
    const float* __restrict__ X, const float* __restrict__ Wrel,
    const float* __restrict__ Wroot, const float* __restrict__ bias,
    float* __restrict__ Yrel, float* __restrict__ Hroot, int nrows)
{
    __shared__ float sW[K * NC];   // interleaved layout

    const int mat = blockIdx.y;
    const float* W    = (mat < NREL) ? (Wrel + (size_t)mat * K * NC) : Wroot;
    float*       Yout = (mat < NREL) ? (Yrel + (size_t)mat * nrows * NC) : Hroot;
    const float* bptr = (mat < NREL) ? nullptr : bias;

    for (int i = threadIdx.x; i < K * NC; i += 256) {
        const int k   = i / NC;          // NC is a power of two -> shifts
        const int col = i & (NC - 1);
        sW[((k >> 1) * NC + col) * 2 + (k & 1)] = W[i];
    }
    __syncthreads();

    const int wave = threadIdx.x >> 5;
    const int lane = threadIdx.x & 31;
    const int half = lane >> 4;     // lanes 16-31 carry K+2..K+3 (A) / M+8 (C)
    const int lm   = lane & 15;

    const int tile_m   = blockIdx.x * 8 + wave;
    const int ntiles_m = nrows >> 4;
    if (tile_m >= ntiles_m) return;

    constexpr int NT = NC >> 4;
    v8f acc[NT] = {};

    const float* xrow = X + (size_t)(tile_m * 16 + lm) * K;

    #pragma unroll 4
    for (int k0 = 0; k0 < K; k0 += 4) {
        const int krow = k0 + 2 * half;            // even
        // A 16x4 f32 fragment: vgpr0 = A[m][krow], vgpr1 = A[m][krow+1]
        const v2f a = *(const v2f*)(xrow + krow);
        #pragma unroll
        for (int t = 0; t < NT; ++t) {
            // B 4x16 fragment: one ds_load_b64 thanks to interleaved layout
            const v2f b = *(const v2f*)(sW + ((krow >> 1) * NC + t * 16 + lm) * 2);
            acc[t] = __builtin_amdgcn_wmma_f32_16x16x4_f32(
                false, a, false, b, (short)0, acc[t], false, false);
        }
    }

    #pragma unroll
    for (int t = 0; t < NT; ++t) {
        const int col = t * 16 + lm;
        const float bv = bptr ? bptr[col] : 0.0f;
        #pragma unroll
        for (int v = 0; v < 8; ++v) {
            const int row = tile_m * 16 + v + 8 * half;   // C/D layout
            Yout[(size_t)row * NC + col] = acc[t][v] + bv;
        }
    }
}

// ---------------------------------------------------------------------------
// Per-relation in-degree counts (stored as float for later division).
// ---------------------------------------------------------------------------
__global__ __launch_bounds__(256) void count_kernel(
    const int* __restrict__ et, const int* __restrict__ ei,
    float* __restrict__ cnt, int E, int n)
{
    const int i = blockIdx.x * 256 + threadIdx.x;
    if (i >= E) return;
    const int r = et[i];
    const int d = ei[E + i];
    unsafeAtomicAdd(&cnt[(size_t)r * n + d], 1.0f);
}

// ---------------------------------------------------------------------------
// Edge scatter: S[r][dst] += Y[r][src]. One thread per float4 of an edge's
// message: global_load_b128 gather + 4x global_atomic_add_f32.
// ---------------------------------------------------------------------------
template <int C>
__global__ __launch_bounds__(256) void scatter_kernel(
    const float* __restrict__ Y, const int* __restrict__ et,
    const int* __restrict__ ei, float* __restrict__ S, int E, int n)
{
    constexpr int Q  = C / 4;                 // float4 chunks per edge
    constexpr int SH = (Q == 16) ? 4 : 3;
    const long long gid = (long long)blockIdx.x * 256 + threadIdx.x;
    const int e = (int)(gid >> SH);
    const int q = ((int)gid & (Q - 1)) * 4;
    if (e >= E) return;
    const int r = et[e];
    const int s = ei[e];
    const int d = ei[E + e];
    const v4f val = *(const v4f*)(Y + ((size_t)r * n + s) * C + q);
    float* dp = S + ((size_t)r * n + d) * C + q;
    unsafeAtomicAdd(dp + 0, val.x);
    unsafeAtomicAdd(dp + 1, val.y);
    unsafeAtomicAdd(dp + 2, val.z);
    unsafeAtomicAdd(dp + 3, val.w);
}

// ---------------------------------------------------------------------------
// z = (relu?)(H + sum_r S_r / max(cnt_r, 1)), written in place over H.
// One thread per float4.
// ---------------------------------------------------------------------------
template <int C, bool RELU>
__global__ __launch_bounds__(256) void finalize_kernel(
    float* __restrict__ H, const float* __restrict__ S,
    const float* __restrict__ cnt, int n)
{
    constexpr int SH = (C == 64) ? 6 : 5;
    const long long i4 = (long long)blockIdx.x * 256 + threadIdx.x;
    if (i4 >= (long long)n * C / 4) return;
    const long long i = i4 * 4;
    const int node = (int)(i >> SH);
    v4f acc = *(const v4f*)(H + i);
    #pragma unroll
    for (int r = 0; r < NREL; ++r) {
        const float cv = fmaxf(cnt[(size_t)r * n + node], 1.0f);
        const v4f sv = *(const v4f*)(S + (size_t)r * n * C + i);
        acc.x += sv.x / cv;
        acc.y += sv.y / cv;
        acc.z += sv.z / cv;
        acc.w += sv.w / cv;
    }
    if (RELU) {
        acc.x = fmaxf(acc.x, 0.0f);
        acc.y = fmaxf(acc.y, 0.0f);
        acc.z = fmaxf(acc.z, 0.0f);
        acc.w = fmaxf(acc.w, 0.0f);
    }
    *(v4f*)(H + i) = acc;
}

// ---------------------------------------------------------------------------
// Decoder: per 16-edge tile, h = relu([z2[s]|z2[d]] @ Wd1 + bd1) via WMMA,
// then out = h @ Wd2 + bd2 via per-lane dot + 16-lane shfl_xor reduction.
// Wd1 staged in LDS pair-interleaved like the GEMM kernel.
// ---------------------------------------------------------------------------
__global__ __launch_bounds__(256) void decoder_kernel(
    const float* __restrict__ z2, const int* __restrict__ pred,
    const float* __restrict__ Wd1, const float* __restrict__ bd1,
    const float* __restrict__ Wd2, const float* __restrict__ bd2,
    float* __restrict__ out, int P)
{
    __shared__ float sW[64 * 64];   // interleaved
    __shared__ float sb1[64];
    __shared__ float sw2[64];
    for (int i = threadIdx.x; i < 64 * 64; i += 256) {
        const int k   = i >> 6;
        const int col = i & 63;
        sW[((k >> 1) * 64 + col) * 2 + (k & 1)] = Wd1[i];
    }
    if (threadIdx.x < 64) {
        sb1[threadIdx.x] = bd1[threadIdx.x];
        sw2[threadIdx.x] = Wd2[threadIdx.x];
    }
    __syncthreads();

    const int wave = threadIdx.x >> 5;
    const int lane = threadIdx.x & 31;
    const int half = lane >> 4;
    const int lm   = lane & 15;

    const int tile = blockIdx.x * 8 + wave;
    if (tile * 16 >= P) return;

    const int edge = tile * 16 + lm;
    const int s = pred[edge];
    const int d = pred[P + edge];
    const float* zs = z2 + (size_t)s * OUTC;
    const float* zd = z2 + (size_t)d * OUTC;

    v8f acc[4] = {};
    #pragma unroll
    for (int step = 0; step < 16; ++step) {
        const int keff = step * 4 + 2 * half;            // 0..62, even
        const float* base = (keff < 32) ? (zs + keff) : (zd + (keff - 32));
        const v2f a = *(const v2f*)base;
        #pragma unroll
        for (int t = 0; t < 4; ++t) {
            const v2f b = *(const v2f*)(sW + ((keff >> 1) * 64 + t * 16 + lm) * 2);
            acc[t] = __builtin_amdgcn_wmma_f32_16x16x4_f32(
                false, a, false, b, (short)0, acc[t], false, false);
        }
    }

    const float b2v = bd2[0];
    #pragma unroll
    for (int v = 0; v < 8; ++v) {
        float p = 0.0f;
        #pragma unroll
        for (int t = 0; t < 4; ++t) {
            const int col = t * 16 + lm;
            float h = acc[t][v] + sb1[col];
            h = h > 0.0f ? h : 0.0f;
            p += h * sw2[col];
        }
        // reduce across the 16 lanes of this half-wave
        p += __shfl_xor(p, 1, 32);
        p += __shfl_xor(p, 2, 32);
        p += __shfl_xor(p, 4, 32);
        p += __shfl_xor(p, 8, 32);
        if (lm == 0) {
            const int row = tile * 16 + v + 8 * half;
            if (row < P) out[row] = p + b2v;
        }
    }
}

// ---------------------------------------------------------------------------
extern "C" void kernel_launch(void* const* d_in, const int* in_sizes, int n_in,
                              void* d_out, int out_size, void* d_ws, size_t ws_size,
                              hipStream_t stream) {
    const float* x     = (const float*)d_in[0];
    const int*   ei    = (const int*)  d_in[1];   // [2, E]: row0=src, row1=dst
    const int*   et    = (const int*)  d_in[2];   // [E]
    const int*   pred  = (const int*)  d_in[3];   // [2, P]
    const float* W1    = (const float*)d_in[4];   // [3,128,64]
    const float* root1 = (const float*)d_in[5];   // [128,64]
    const float* b1    = (const float*)d_in[6];   // [64]
    const float* W2    = (const float*)d_in[7];   // [3,64,32]
    const float* root2 = (const float*)d_in[8];   // [64,32]
    const float* b2    = (const float*)d_in[9];   // [32]
    const float* Wd1   = (const float*)d_in[10];  // [64,64]
    const float* bd1   = (const float*)d_in[11];  // [64]
    const float* Wd2   = (const float*)d_in[12];  // [64,1]
    const float* bd2   = (const float*)d_in[13];  // [1]
    float* out = (float*)d_out;

    // Workspace layout (bytes); regions reused across layers (stream-ordered).
    char* ws = (char*)d_ws;
    float* Y1  = (float*)(ws + 0);             // 3*N*64 f32 = 76.8 MB (reused as Y2)
    float* S1  = (float*)(ws + 76800000);      // 3*N*64 f32 = 76.8 MB (reused as S2)
    float* Z1  = (float*)(ws + 153600000);     // N*64 f32   = 25.6 MB (H1 -> z1 in place)
    float* Z2  = (float*)(ws + 179200000);     // N*32 f32   = 12.8 MB (H2 -> z2 in place)
    float* CNT = (float*)(ws + 192000000);     // 3*N f32    =  1.2 MB
    float* Y2 = Y1;
    float* S2 = S1;

    const int ntiles = NNODES / 16;                 // 6250
    const dim3 gemm_grid((ntiles + 7) / 8, 4);      // 8 waves/block, 4 matrices

    // ---- Layer 1 ----
    hipMemsetAsync(S1,  0, (size_t)NREL * NNODES * HIDC * sizeof(float), stream);
    hipMemsetAsync(CNT, 0, (size_t)NREL * NNODES * sizeof(float), stream);

    gemm_wmma_kernel<INC, HIDC><<<gemm_grid, 256, 0, stream>>>(
        x, W1, root1, b1, Y1, Z1, NNODES);
    count_kernel<<<(NEDGES + 255) / 256, 256, 0, stream>>>(et, ei, CNT, NEDGES, NNODES);
    scatter_kernel<HIDC><<<(unsigned)(((long long)NEDGES * (HIDC / 4)) / 256), 256, 0, stream>>>(
        Y1, et, ei, S1, NEDGES, NNODES);
    finalize_kernel<HIDC, true><<<(NNODES * HIDC / 4 + 255) / 256, 256, 0, stream>>>(
        Z1, S1, CNT, NNODES);

    // ---- Layer 2 ----
    hipMemsetAsync(S2, 0, (size_t)NREL * NNODES * OUTC * sizeof(float), stream);

    gemm_wmma_kernel<HIDC, OUTC><<<gemm_grid, 256, 0, stream>>>(
        Z1, W2, root2, b2, Y2, Z2, NNODES);
    scatter_kernel<OUTC><<<(unsigned)(((long long)NEDGES * (OUTC / 4)) / 256), 256, 0, stream>>>(
        Y2, et, ei, S2, NEDGES, NNODES);
    finalize_kernel<OUTC, false><<<(NNODES * OUTC / 4 + 255) / 256, 256, 0, stream>>>(
        Z2, S2, CNT, NNODES);

    // ---- Decoder ----
    const int ptiles = (NPRED + 15) / 16;           // 12500
    decoder_kernel<<<(ptiles + 7) / 8, 256, 0, stream>>>(
        Z2, pred, Wd1, bd1, Wd2, bd2, out, NPRED);
}